// GinNet_64991445123371
// MI455X (gfx1250) — compile-verified
//
#include <hip/hip_runtime.h>

typedef float v2f __attribute__((ext_vector_type(2)));
typedef float v8f __attribute__((ext_vector_type(8)));

// ---------------------------------------------------------------- zero fill
__global__ void zero_f32(float4* __restrict__ p, int n4) {
    int i = blockIdx.x * blockDim.x + threadIdx.x;
    if (i < n4) p[i] = make_float4(0.f, 0.f, 0.f, 0.f);
}

// ---------------------------------------------------------------- edge scatter-add
// agg[dst[e], :] += h[src[e], :]   (F floats per edge, F%4==0)
// 16 (or 4) threads per edge -> coalesced 256B gather, 4 atomics per thread.
__global__ void scatter_add_k(const float* __restrict__ h,
                              const int* __restrict__ src,
                              const int* __restrict__ dst,
                              float* __restrict__ agg,
                              int E, int F4, int F) {
    long tid = (long)blockIdx.x * blockDim.x + threadIdx.x;
    int e = (int)(tid / F4);
    if (e >= E) return;
    int fg = (int)(tid - (long)e * F4) * 4;
    int s = src[e];
    int d = dst[e];
    const float4 v = *(const float4*)(h + (long)s * F + fg);
    float* ap = agg + (long)d * F + fg;
    atomicAdd(ap + 0, v.x);
    atomicAdd(ap + 1, v.y);
    atomicAdd(ap + 2, v.z);
    atomicAdd(ap + 3, v.w);
}

// ---------------------------------------------------------------- fused GIN MLP
// out = ReLU( ReLU( ((1+eps)*h + agg) @ W1 + b1 ) @ W2 + b2 )
// 4 waves per block, one 16-row tile per wave, f32 WMMA (16x16x4).
template <int DIN>
__global__ __launch_bounds__(128) void gin_mlp(const float* __restrict__ h,
                                               const float* __restrict__ agg,
                                               const float* __restrict__ eps_p,
                                               const float* __restrict__ w1,
                                               const float* __restrict__ b1,
                                               const float* __restrict__ w2,
                                               const float* __restrict__ b2,
                                               float* __restrict__ out,
                                               int n_nodes) {
    __shared__ float sW1[DIN * 64];
    __shared__ float sW2[64 * 64];
    __shared__ float sB1[64];
    __shared__ float sB2[64];
    __shared__ float sX[4][16 * 64];   // X tile (16 x DIN, stride 64), later Y1 (16 x 64)

    const int tid  = threadIdx.x;
    const int wave = tid >> 5;
    const int lane = tid & 31;

    for (int i = tid; i < DIN * 64; i += 128) sW1[i] = w1[i];
    for (int i = tid; i < 64 * 64; i += 128)  sW2[i] = w2[i];
    if (tid < 64) { sB1[tid] = b1[tid]; sB2[tid] = b2[tid]; }
    const float eps1 = 1.0f + eps_p[0];

    const int rowBase = blockIdx.x * 64 + wave * 16;

    // stage X = (1+eps)*h + agg into own wave's LDS tile
    for (int i = lane; i < 16 * DIN; i += 32) {
        int r = i / DIN, c = i - r * DIN;
        int row = rowBase + r;
        float v = 0.0f;
        if (row < n_nodes) v = eps1 * h[(long)row * DIN + c] + agg[(long)row * DIN + c];
        sX[wave][r * 64 + c] = v;
    }
    __syncthreads();

    const int hl = lane >> 4;   // half: 0 -> K {k0,k0+1}, 1 -> K {k0+2,k0+3}
    const int r  = lane & 15;

    // ---- GEMM1: X(16xDIN) @ W1(DINx64) ----
    v8f acc0 = {0,0,0,0,0,0,0,0};
    v8f acc[4];
#pragma unroll
    for (int n = 0; n < 4; ++n) acc[n] = acc0;

    for (int kk = 0; kk < DIN / 4; ++kk) {
        int k0 = kk * 4 + 2 * hl;
        v2f af;
        af.x = sX[wave][r * 64 + k0];
        af.y = sX[wave][r * 64 + k0 + 1];
#pragma unroll
        for (int n = 0; n < 4; ++n) {
            v2f bf;
            bf.x = sW1[k0 * 64 + n * 16 + r];
            bf.y = sW1[(k0 + 1) * 64 + n * 16 + r];
            acc[n] = __builtin_amdgcn_wmma_f32_16x16x4_f32(
                false, af, false, bf, (short)0, acc[n], false, false);
        }
    }

    // bias + ReLU, write Y1 over the X tile (DS ops in-order per wave -> safe)
#pragma unroll
    for (int n = 0; n < 4; ++n) {
        int col = n * 16 + r;
        float bias = sB1[col];
#pragma unroll
        for (int j = 0; j < 8; ++j) {
            float v = acc[n][j] + bias;
            v = fmaxf(v, 0.0f);
            sX[wave][(j + 8 * hl) * 64 + col] = v;
        }
    }
    __syncthreads();

    // ---- GEMM2: Y1(16x64) @ W2(64x64) ----
#pragma unroll
    for (int n = 0; n < 4; ++n) acc[n] = acc0;

    for (int kk = 0; kk < 16; ++kk) {
        int k0 = kk * 4 + 2 * hl;
        v2f af;
        af.x = sX[wave][r * 64 + k0];
        af.y = sX[wave][r * 64 + k0 + 1];
#pragma unroll
        for (int n = 0; n < 4; ++n) {
            v2f bf;
            bf.x = sW2[k0 * 64 + n * 16 + r];
            bf.y = sW2[(k0 + 1) * 64 + n * 16 + r];
            acc[n] = __builtin_amdgcn_wmma_f32_16x16x4_f32(
                false, af, false, bf, (short)0, acc[n], false, false);
        }
    }

#pragma unroll
    for (int n = 0; n < 4; ++n) {
        int col = n * 16 + r;
        float bias = sB2[col];
#pragma unroll
        for (int j = 0; j < 8; ++j) {
            int row = rowBase + j + 8 * hl;
            if (row < n_nodes) {
                float v = acc[n][j] + bias;
                out[(long)row * 64 + col] = fmaxf(v, 0.0f);
            }
        }
    }
}

// ---------------------------------------------------------------- global add pool
__global__ void pool_k(const float* __restrict__ h, const int* __restrict__ batch,
                       float* __restrict__ g, int N) {
    long tid = (long)blockIdx.x * blockDim.x + threadIdx.x;
    int n = (int)(tid >> 4);
    if (n >= N) return;
    int f = (int)(tid & 15) * 4;
    const float4 v = *(const float4*)(h + (long)n * 64 + f);
    float* gp = g + (long)batch[n] * 64 + f;
    atomicAdd(gp + 0, v.x);
    atomicAdd(gp + 1, v.y);
    atomicAdd(gp + 2, v.z);
    atomicAdd(gp + 3, v.w);
}

// ---------------------------------------------------------------- readout head
// out[g] = ReLU(g@fc1+b1) @ fc2 + b2 ; one wave32 per graph, lane = hidden unit.
__global__ __launch_bounds__(256) void head_k(const float* __restrict__ g,
                                              const float* __restrict__ fc1w,
                                              const float* __restrict__ fc1b,
                                              const float* __restrict__ fc2w,
                                              const float* __restrict__ fc2b,
                                              float* __restrict__ out, int G) {
    int wave = threadIdx.x >> 5;
    int lane = threadIdx.x & 31;
    int gi = blockIdx.x * 8 + wave;
    if (gi >= G) return;
    const float* gr = g + (long)gi * 64;
    float hsum = fc1b[lane];
    for (int k = 0; k < 64; ++k) hsum += gr[k] * fc1w[k * 32 + lane];
    hsum = fmaxf(hsum, 0.0f);
    float v = hsum * fc2w[lane];
    for (int off = 16; off > 0; off >>= 1) v += __shfl_xor(v, off, 32);
    if (lane == 0) out[gi] = v + fc2b[0];
}

// ---------------------------------------------------------------- launcher
extern "C" void kernel_launch(void* const* d_in, const int* in_sizes, int n_in,
                              void* d_out, int out_size, void* d_ws, size_t ws_size,
                              hipStream_t stream) {
    const float* x     = (const float*)d_in[0];
    const int*   ei    = (const int*)d_in[1];   // [0..E) = src, [E..2E) = dst
    const int*   batch = (const int*)d_in[2];
    const int E = in_sizes[1] / 2;
    const int N = in_sizes[2];
    const int G = out_size;

    const float *eps[5], *w1[5], *b1[5], *w2[5], *b2[5];
    int idx = 3;
    for (int i = 0; i < 5; ++i) {
        eps[i] = (const float*)d_in[idx++];
        w1[i]  = (const float*)d_in[idx++];
        b1[i]  = (const float*)d_in[idx++];
        w2[i]  = (const float*)d_in[idx++];
        b2[i]  = (const float*)d_in[idx++];
    }
    const float* fc1w = (const float*)d_in[idx++];
    const float* fc1b = (const float*)d_in[idx++];
    const float* fc2w = (const float*)d_in[idx++];
    const float* fc2b = (const float*)d_in[idx++];

    float* agg  = (float*)d_ws;
    float* hA   = agg + (size_t)N * 64;
    float* hB   = hA + (size_t)N * 64;
    float* gbuf = hB + (size_t)N * 64;

    const float* hin = x;
    float* cur = hA;
    float* nxt = hB;

    for (int layer = 0; layer < 5; ++layer) {
        const int F = (layer == 0) ? 16 : 64;
        // zero agg
        int n4 = (N * F) / 4;
        zero_f32<<<(n4 + 255) / 256, 256, 0, stream>>>((float4*)agg, n4);
        // scatter-add edge messages
        int F4 = F / 4;
        long total = (long)E * F4;
        int sblocks = (int)((total + 255) / 256);
        scatter_add_k<<<sblocks, 256, 0, stream>>>(hin, ei, ei + E, agg, E, F4, F);
        // fused MLP with f32 WMMA
        int mblocks = (N + 63) / 64;
        if (F == 16)
            gin_mlp<16><<<mblocks, 128, 0, stream>>>(hin, agg, eps[layer], w1[layer],
                                                     b1[layer], w2[layer], b2[layer], cur, N);
        else
            gin_mlp<64><<<mblocks, 128, 0, stream>>>(hin, agg, eps[layer], w1[layer],
                                                     b1[layer], w2[layer], b2[layer], cur, N);
        hin = cur;
        float* t = cur; cur = nxt; nxt = t;
    }

    // global add pool -> [G, 64]
    int g4 = (G * 64) / 4;
    zero_f32<<<(g4 + 255) / 256, 256, 0, stream>>>((float4*)gbuf, g4);
    long ptotal = (long)N * 16;
    pool_k<<<(int)((ptotal + 255) / 256), 256, 0, stream>>>(hin, batch, gbuf, N);

    // readout head
    head_k<<<(G + 7) / 8, 256, 0, stream>>>(gbuf, fc1w, fc1b, fc2w, fc2b, (float*)d_out, G);
}